// TorchLMHeadGRPO_24876450578519
// MI455X (gfx1250) — compile-verified
//
#include <hip/hip_runtime.h>

// ---------------------------------------------------------------------------
// GRPO fused LM-head loss for MI455X (gfx1250, wave32).
//   logits = x @ W^T ;  ptl = logit[sel] - logsumexp(logits)
//   loss   = mean_mask( -adv + BETA*(exp(d)-d-1) ),  d = ref_ptl - ptl
//   coef_1 == exp(0) == 1  =>  clip metric is exactly 0.
// Bandwidth-bound on W (524 MB @ 23.3 TB/s ~ 22.5 us). fp32 WMMA
// (V_WMMA_F32_16X16X4_F32) keeps reference numerics; W tiles are streamed
// into double-buffered LDS by the Tensor Data Mover (TENSORcnt) and overlap
// with the WMMA pipeline.
// ---------------------------------------------------------------------------

typedef __attribute__((ext_vector_type(2))) float        v2f;
typedef __attribute__((ext_vector_type(8))) float        v8f;
typedef __attribute__((ext_vector_type(4))) unsigned int v4u;
typedef __attribute__((ext_vector_type(4))) int          v4i;
typedef __attribute__((ext_vector_type(8))) int          v8i;

#define TOK   2048                 // B*S
#define SEQ   512
#define HDIM  2048
#define VDIM  32000
#define MTILE 16
#define NTILE 128                  // 8 waves * 16 columns
#define KTILE 64
#define LDS_STRIDE (KTILE + 4)     // +4 DWORD pad per row (banks); TDM pads too
#define VCHUNKS 10
#define TILES_PER_CHUNK 25         // 25 * 128 * 10 = 32000
#define KCHUNKS (HDIM / KTILE)     // 32
#define TOKEN_TILES (TOK / MTILE)  // 128
#define GBETA 0.1f

#if __has_builtin(__builtin_amdgcn_tensor_load_to_lds) && \
    __has_builtin(__builtin_amdgcn_s_wait_tensorcnt)
#define USE_TDM 1
#else
#define USE_TDM 0
#endif

#if USE_TDM
typedef __attribute__((address_space(3))) void lds_void_t;

__device__ __forceinline__ unsigned lds_byte_off(void* p) {
    return (unsigned)(unsigned long long)(lds_void_t*)p;
}

// Issue one TDM descriptor: 2D tile (KTILE x NTILE) of f32 from global (row
// stride HDIM elements) into LDS at ldsDst, with 4-DWORD padding inserted
// every 64 DWORDs (matches LDS_STRIDE = 68 floats).
__device__ __forceinline__ void tdm_load_w_tile(const float* gsrc, unsigned ldsDst) {
    unsigned long long ga = (unsigned long long)(size_t)gsrc;
    v4u g0;
    g0.x = 1u;                                             // count = 1 (valid)
    g0.y = ldsDst;                                         // lds_addr (bytes)
    g0.z = (unsigned)ga;                                   // global_addr[31:0]
    g0.w = (unsigned)((ga >> 32) & 0x01FFFFFFu)            // global_addr[56:32]
         | (2u << 30);                                     // type = 2 ("image")
    v8i g1;
    g1[0] = (2 << 16)                                      // data_size = 4B
          | (1 << 20)                                      // pad_enable
          | (5 << 22)                                      // pad_interval: 64 DW
          | (3 << 25);                                     // pad_amount:   4 DW
    g1[1] = (KTILE & 0xFFFF) << 16;                        // tensor_dim0[15:0]
    g1[2] = (NTILE & 0xFFFF) << 16;                        // tensor_dim1[15:0]
    g1[3] = (KTILE & 0xFFFF) << 16;                        // tile_dim0
    g1[4] = NTILE;                                         // tile_dim1 (tile_dim2=0)
    g1[5] = HDIM;                                          // tensor_dim0_stride
    g1[6] = 0;
    g1[7] = 0;
    v4i z4 = {0, 0, 0, 0};
#if defined(__clang_major__) && __clang_major__ >= 23
    v8i z8 = {0, 0, 0, 0, 0, 0, 0, 0};
    __builtin_amdgcn_tensor_load_to_lds(g0, g1, z4, z4, z8, 0);
#else
    __builtin_amdgcn_tensor_load_to_lds(g0, g1, z4, z4, 0);
#endif
}
#endif // USE_TDM

__global__ __launch_bounds__(256)
void grpo_lse_kernel(const float* __restrict__ Xp, const float* __restrict__ Wp,
                     const float* __restrict__ Xr, const float* __restrict__ Wr,
                     const int*   __restrict__ ids,
                     float* __restrict__ sel, float2* __restrict__ part)
{
    __shared__ float As[2][MTILE][LDS_STRIDE];
    __shared__ float Ws[2][NTILE][LDS_STRIDE];
    __shared__ float redM[8][MTILE];
    __shared__ float redS[8][MTILE];
    __shared__ int   ids_s[MTILE];

    const int tid  = threadIdx.x;
    const int wave = tid >> 5;
    const int lane = tid & 31;
    const int half = lane >> 4;   // K-split half (A/B frags) / M-split half (C)
    const int l    = lane & 15;

    const int tt0    = blockIdx.x * MTILE;
    const int vcBase = blockIdx.y * (TILES_PER_CHUNK * NTILE);
    const int mat    = blockIdx.z;

    const float* __restrict__ X  = mat ? Xr : Xp;
    const float* __restrict__ Wm = mat ? Wr : Wp;

    if (tid < MTILE) ids_s[tid] = ids[tt0 + tid];

    const int arow = tid >> 4;          // A staging: 16 x 64 floats, 1 float4/thr
    const int acol = (tid & 15) * 4;

#if USE_TDM
    unsigned wsBase[2];
    wsBase[0] = lds_byte_off(&Ws[0][0][0]);
    wsBase[1] = lds_byte_off(&Ws[1][0][0]);
#endif

    // stage chunk g into LDS buffer buf (A: all threads, W: TDM or manual)
    auto stage = [&](int g, int buf) {
        const int tile = g >> 5;
        const int k0   = (g & 31) * KTILE;
        const int n0   = vcBase + tile * NTILE;
        *(float4*)&As[buf][arow][acol] =
            *(const float4*)&X[(size_t)(tt0 + arow) * HDIM + k0 + acol];
#if USE_TDM
        if (wave == 0)
            tdm_load_w_tile(&Wm[(size_t)n0 * HDIM + k0], wsBase[buf]);
#else
        float4 tmp[8];                 // register-stage: batch 8 loads, 1 wait
#pragma unroll
        for (int j = 0; j < 8; ++j) {
            const int flat = tid + j * 256;
            tmp[j] = *(const float4*)&Wm[(size_t)(n0 + (flat >> 4)) * HDIM +
                                         k0 + (flat & 15) * 4];
        }
#pragma unroll
        for (int j = 0; j < 8; ++j) {
            const int flat = tid + j * 256;
            *(float4*)&Ws[buf][flat >> 4][(flat & 15) * 4] = tmp[j];
        }
#endif
    };

    float rmax[8], rsum[8];             // online softmax stats (per wave slice)
#pragma unroll
    for (int r = 0; r < 8; ++r) { rmax[r] = -3.0e38f; rsum[r] = 0.f; }

    v8f c = {0.f,0.f,0.f,0.f,0.f,0.f,0.f,0.f};

    const int NG = TILES_PER_CHUNK * KCHUNKS;   // 800 chunks per block
    stage(0, 0);

    for (int g = 0; g < NG; ++g) {
        const int buf = g & 1;
#if USE_TDM
        __builtin_amdgcn_s_wait_tensorcnt(0);   // TDM(g) landed (waves 1-7: cnt==0)
#endif
        __syncthreads();                        // buf ready; buf^1 readers done
        if (g + 1 < NG) stage(g + 1, buf ^ 1);  // DMA overlaps WMMA below

        // 16 x V_WMMA_F32_16X16X4_F32 per chunk per wave.
        // A frag (16x4 f32): lanes 0-15 K={kk,kk+1}, lanes 16-31 K={kk+2,kk+3}
        // B frag (4x16 f32): same K split, N = lane&15 (wave's vocab rows)
#pragma unroll
        for (int kk = 0; kk < KTILE; kk += 4) {
            const int colk = kk + 2 * half;
            v2f a = *(const v2f*)&As[buf][l][colk];
            v2f b = *(const v2f*)&Ws[buf][(wave << 4) + l][colk];
            c = __builtin_amdgcn_wmma_f32_16x16x4_f32(
                    false, a, false, b, (short)0, c, false, false);
        }

        if ((g & 31) == 31) {                   // end of one 128-col vocab tile
            const int n0   = vcBase + (g >> 5) * NTILE;
            const int vcol = n0 + (wave << 4) + l;
            // C/D layout: c[r] of lane = (token m = r + 8*half, col = l)
#pragma unroll
            for (int r = 0; r < 8; ++r) {
                const int m = r + 8 * half;
                if (ids_s[m] == vcol)           // unique writer per token
                    sel[mat * TOK + tt0 + m] = c[r];

                float tmax = c[r];
#pragma unroll
                for (int s = 1; s < 16; s <<= 1)
                    tmax = fmaxf(tmax, __shfl_xor(tmax, s));
                float e = __expf(c[r] - tmax);
#pragma unroll
                for (int s = 1; s < 16; s <<= 1)
                    e += __shfl_xor(e, s);

                const float nm = fmaxf(rmax[r], tmax);
                rsum[r] = rsum[r] * __expf(rmax[r] - nm) + e * __expf(tmax - nm);
                rmax[r] = nm;
            }
            c = (v8f){0.f,0.f,0.f,0.f,0.f,0.f,0.f,0.f};
        }
    }

    // merge the 8 waves' stats -> one (max,sum) per token per block
    if (l == 0) {
#pragma unroll
        for (int r = 0; r < 8; ++r) {
            redM[wave][r + 8 * half] = rmax[r];
            redS[wave][r + 8 * half] = rsum[r];
        }
    }
    __syncthreads();
    if (tid < MTILE) {
        float M = -3.0e38f, S = 0.f;
#pragma unroll
        for (int w = 0; w < 8; ++w) {
            const float m2 = redM[w][tid], s2 = redS[w][tid];
            const float nm = fmaxf(M, m2);
            S = S * __expf(M - nm) + s2 * __expf(m2 - nm);
            M = nm;
        }
        part[((size_t)(mat * TOKEN_TILES + blockIdx.x) * VCHUNKS + blockIdx.y) * MTILE + tid]
            = make_float2(M, S);
    }
}

__global__ __launch_bounds__(256)
void grpo_finalize_kernel(const float*  __restrict__ sel,
                          const float2* __restrict__ part,
                          const float*  __restrict__ amask,
                          const float*  __restrict__ adv,
                          float* __restrict__ out)
{
    const int tid = threadIdx.x;
    float lsum = 0.f, ksum = 0.f, msum = 0.f;

    for (int t = tid; t < TOK; t += 256) {
        const int tt = t >> 4, m = t & 15;
        float lp[2];
#pragma unroll
        for (int mat = 0; mat < 2; ++mat) {
            float M = -3.0e38f, S = 0.f;
            for (int vc = 0; vc < VCHUNKS; ++vc) {
                const float2 p =
                    part[((size_t)(mat * TOKEN_TILES + tt) * VCHUNKS + vc) * MTILE + m];
                const float nm = fmaxf(M, p.x);
                S = S * __expf(M - nm) + p.y * __expf(p.x - nm);
                M = nm;
            }
            lp[mat] = sel[mat * TOK + t] - (__logf(S) + M);
        }
        const float d  = lp[1] - lp[0];
        const float kl = __expf(d) - d - 1.f;
        const float mk = amask[t];
        lsum += (-adv[t / SEQ] + GBETA * kl) * mk;   // coef_1 == 1 identically
        ksum += kl * mk;
        msum += mk;
    }

    __shared__ float red[3][256];
    red[0][tid] = lsum; red[1][tid] = ksum; red[2][tid] = msum;
    __syncthreads();
    for (int s = 128; s > 0; s >>= 1) {
        if (tid < s) {
            red[0][tid] += red[0][tid + s];
            red[1][tid] += red[1][tid + s];
            red[2][tid] += red[2][tid + s];
        }
        __syncthreads();
    }
    if (tid == 0) {
        const float denom = fmaxf(red[2][0], 1.f);
        out[0] = red[0][0] / denom;   // loss
        out[1] = red[1][0] / denom;   // kl metric
        out[2] = 0.f;                 // clip metric (coef_1 == 1 -> never clipped)
    }
}

extern "C" void kernel_launch(void* const* d_in, const int* in_sizes, int n_in,
                              void* d_out, int out_size, void* d_ws, size_t ws_size,
                              hipStream_t stream) {
    const float* x    = (const float*)d_in[0];
    const float* W    = (const float*)d_in[1];
    const float* rx   = (const float*)d_in[2];
    const float* rW   = (const float*)d_in[3];
    const int*   ids  = (const int*)  d_in[4];
    const float* mask = (const float*)d_in[5];
    const float* adv  = (const float*)d_in[6];
    float* out = (float*)d_out;

    // workspace: [2*TOK] selected logits, then [2*128*10*16] (max,sum) partials
    float*  sel  = (float*)d_ws;
    float2* part = (float2*)(sel + 2 * TOK);

    dim3 grid(TOKEN_TILES, VCHUNKS, 2);
    grpo_lse_kernel<<<grid, 256, 0, stream>>>(x, W, rx, rW, ids, sel, part);
    grpo_finalize_kernel<<<1, 256, 0, stream>>>(sel, part, mask, adv, out);
}